// GConvLayer_59330678227073
// MI455X (gfx1250) — compile-verified
//
#include <hip/hip_runtime.h>
#include <hip/hip_bf16.h>

#define HIDDEN 128

typedef __attribute__((ext_vector_type(2))) float v2f;
typedef __attribute__((ext_vector_type(8))) float v8f;

// ---------------------------------------------------------------------------
// Kernel 1: m = relu(x @ W^T + b) via V_WMMA_F32_16X16X4_F32 (wave32).
// One wave -> one 16x16 output tile. 8 waves/block -> 16-row x 128-col strip.
// ---------------------------------------------------------------------------
__global__ __launch_bounds__(256) void gconv_gemm_relu(
    const float* __restrict__ x, const float* __restrict__ W,
    const float* __restrict__ bias, float* __restrict__ m, int N) {
  const int lane = threadIdx.x & 31;
  const int wave = threadIdx.x >> 5;            // 0..7 -> column tile
  const int row_base = blockIdx.x * 16;
  const int col_base = wave * 16;
  if (row_base >= N) return;                    // block-uniform: EXEC stays full

  const int n    = lane & 15;                   // M index for A, N index for B/D
  const int kgrp = (lane >> 4) * 2;             // half-wave K offset (0 or 2)

  const float* __restrict__ xrow = x + (size_t)(row_base + n) * HIDDEN;
  const float* __restrict__ wrow = W + (size_t)(col_base + n) * HIDDEN;

  v8f acc = {};
#pragma unroll
  for (int k = 0; k < HIDDEN; k += 4) {
    v2f a = *(const v2f*)(xrow + k + kgrp);     // A[16x4]: lane n, K=kgrp..kgrp+1
    v2f b = *(const v2f*)(wrow + k + kgrp);     // B[4x16]=W^T: K=kgrp..kgrp+1, col n
    acc = __builtin_amdgcn_wmma_f32_16x16x4_f32(
        /*neg_a=*/false, a, /*neg_b=*/false, b,
        /*c_mod=*/(short)0, acc, /*reuse_a=*/false, /*reuse_b=*/false);
  }

  const float bv = bias[col_base + n];
  const int moff = (lane >> 4) * 8;             // D: VGPR g -> row g (+8 upper half)
  float* __restrict__ mp = m + (size_t)row_base * HIDDEN + col_base + n;
#pragma unroll
  for (int g = 0; g < 8; ++g) {
    float v = acc[g] + bv;
    mp[(size_t)(g + moff) * HIDDEN] = v > 0.0f ? v : 0.0f;
  }
}

// ---------------------------------------------------------------------------
// Kernel 0: zero agg + deg (workspace is not re-poisoned between replays).
// ---------------------------------------------------------------------------
__global__ void gconv_zero(float* __restrict__ p, size_t n) {
  size_t i = (size_t)blockIdx.x * blockDim.x + threadIdx.x;
  if (i < n) p[i] = 0.0f;
}

// ---------------------------------------------------------------------------
// Kernel 2: edge scatter. 128 threads per edge: 512B coalesced gather of
// m[cols[e]] + 512B of contiguous f32 atomics into agg[rows[e]] (L2-resident:
// m and agg are 51.2MB each, both fit in the 192MB L2). Lane f==0 also counts
// the degree.
// ---------------------------------------------------------------------------
__global__ __launch_bounds__(256) void gconv_scatter(
    const float* __restrict__ m, const int* __restrict__ rows,
    const int* __restrict__ cols, float* __restrict__ agg,
    float* __restrict__ deg, int nEdges) {
  long long t = (long long)blockIdx.x * blockDim.x + threadIdx.x;
  int e = (int)(t >> 7);
  if (e >= nEdges) return;
  int f = (int)(t & (HIDDEN - 1));
  int r = rows[e];
  int c = cols[e];
  atomicAdd(&agg[(size_t)r * HIDDEN + f], m[(size_t)c * HIDDEN + f]);
  if (f == 0) atomicAdd(&deg[r], 1.0f);
}

// ---------------------------------------------------------------------------
// Kernel 3: per-node finalize: h = x + agg/max(d,1); RMSNorm(h)*gamma + beta.
// One wave32 per node, 4 floats per lane, shuffle-reduce the sum of squares.
// ---------------------------------------------------------------------------
__global__ __launch_bounds__(256) void gconv_finalize(
    const float* __restrict__ x, const float* __restrict__ agg,
    const float* __restrict__ deg, const float* __restrict__ gamma,
    const float* __restrict__ beta, float* __restrict__ out, int N) {
  int node = blockIdx.x * (blockDim.x >> 5) + (threadIdx.x >> 5);
  if (node >= N) return;
  const int lane = threadIdx.x & 31;

  float d = deg[node];
  float invd = 1.0f / (d == 0.0f ? 1.0f : d);

  const float4 xv = *(const float4*)(x + (size_t)node * HIDDEN + lane * 4);
  const float4 av = *(const float4*)(agg + (size_t)node * HIDDEN + lane * 4);
  float4 h;
  h.x = xv.x + av.x * invd;
  h.y = xv.y + av.y * invd;
  h.z = xv.z + av.z * invd;
  h.w = xv.w + av.w * invd;

  float ss = h.x * h.x + h.y * h.y + h.z * h.z + h.w * h.w;
#pragma unroll
  for (int off = 16; off >= 1; off >>= 1) ss += __shfl_xor(ss, off, 32);
  const float inv_rms = rsqrtf(ss * (1.0f / HIDDEN) + 1e-5f);

  const float4 gv = *(const float4*)(gamma + lane * 4);
  const float4 bv = *(const float4*)(beta + lane * 4);
  float4 o;
  o.x = h.x * inv_rms * gv.x + bv.x;
  o.y = h.y * inv_rms * gv.y + bv.y;
  o.z = h.z * inv_rms * gv.z + bv.z;
  o.w = h.w * inv_rms * gv.w + bv.w;
  *(float4*)(out + (size_t)node * HIDDEN + lane * 4) = o;
}

extern "C" void kernel_launch(void* const* d_in, const int* in_sizes, int n_in,
                              void* d_out, int out_size, void* d_ws, size_t ws_size,
                              hipStream_t stream) {
  const float* x     = (const float*)d_in[0];
  const float* W     = (const float*)d_in[1];
  const float* bias  = (const float*)d_in[2];
  const float* gamma = (const float*)d_in[3];
  const float* beta  = (const float*)d_in[4];
  const int*   rows  = (const int*)d_in[5];
  const int*   cols  = (const int*)d_in[6];
  float*       out   = (float*)d_out;

  const int N      = in_sizes[0] / HIDDEN;   // 100000
  const int nEdges = in_sizes[5];            // 1600000

  float* m   = (float*)d_ws;                 // [N, 128]  51.2 MB
  float* agg = m + (size_t)N * HIDDEN;       // [N, 128]  51.2 MB
  float* deg = agg + (size_t)N * HIDDEN;     // [N]        0.4 MB

  // zero agg + deg (contiguous)
  size_t zn = (size_t)N * HIDDEN + (size_t)N;
  gconv_zero<<<(int)((zn + 255) / 256), 256, 0, stream>>>(agg, zn);

  // m = relu(x @ W^T + b): 16 rows per block, 8 column-tile waves per block
  gconv_gemm_relu<<<(N + 15) / 16, 256, 0, stream>>>(x, W, bias, m, N);

  // edge scatter + degree count
  long long tot = (long long)nEdges * HIDDEN;
  gconv_scatter<<<(int)((tot + 255) / 256), 256, 0, stream>>>(m, rows, cols, agg,
                                                              deg, nEdges);

  // residual + RMSNorm, one wave per node (8 nodes per 256-thread block)
  gconv_finalize<<<(N + 7) / 8, 256, 0, stream>>>(x, agg, deg, gamma, beta, out, N);
}